// TFN_78134045049149
// MI455X (gfx1250) — compile-verified
//
#include <hip/hip_runtime.h>
#include <hip/hip_bf16.h>

typedef float v2f __attribute__((ext_vector_type(2)));
typedef float v8f __attribute__((ext_vector_type(8)));

#define ADIM 32
#define VDIM 32
#define TDIM 128
#define PF   128
#define AV   1089   // 33*33
#define KT   129    // T+1
#define BATCH 256
#define SLICES 33
#define AV_PER_SLICE 33
#define MROWS 32    // rows of batch per workgroup (2 M-tiles per wave)

// t_aug element: k==0 -> 1, 1..128 -> text[m][k-1], >=129 -> 0 (pad)
__device__ __forceinline__ float taug_elem(const float* __restrict__ text, int m, int k) {
    if (k == 0) return 1.0f;
    if (k <= TDIM) return text[m * TDIM + (k - 1)];
    return 0.0f;
}

// Kernel 1: hpre[m][n] += sum over avi-slice of (a_aug*v_aug)[m,avi] * (t_aug[m,:] @ W1[avi])
// Each wave owns a 32x16 C slab (two 16x16 tiles sharing every B fragment -> 1 load : 1 WMMA,
// two independent accumulation chains). A-frags (t_aug) live in 132 persistent VGPRs.
__global__ __launch_bounds__(256) void tfn_w1_kernel(
    const float* __restrict__ audio, const float* __restrict__ video,
    const float* __restrict__ text,  const float* __restrict__ W1,
    float* __restrict__ hpre)
{
    __shared__ float av_lds[MROWS * AV_PER_SLICE];  // slice-local (a*v) table: 4.2 KB

    const int tid  = threadIdx.x;
    const int m0   = blockIdx.x * MROWS;        // batch-row base (32 rows)
    const int avi0 = blockIdx.y * AV_PER_SLICE; // avi slice base
    const int lane = tid & 31;
    const int wave = tid >> 5;
    const int lm   = lane & 15;        // A: M index / B,C: N index within tile
    const int hi   = lane >> 4;        // half-wave select
    const int kb   = hi * 2;           // K sub-offset {0,2}
    const int nb   = wave * 16;        // N-tile base (8 waves cover N=128)

    // ---- build slice-local (a_aug * v_aug) table in LDS: 32 rows x 33 avi ----
    for (int idx = tid; idx < MROWS * AV_PER_SLICE; idx += 256) {
        int mm  = idx / AV_PER_SLICE;
        int j   = idx - mm * AV_PER_SLICE;
        int avi = avi0 + j;
        int ai  = avi / 33;
        int vi  = avi - ai * 33;
        float av = (ai == 0) ? 1.0f : audio[(m0 + mm) * ADIM + (ai - 1)];
        float vv = (vi == 0) ? 1.0f : video[(m0 + mm) * VDIM + (vi - 1)];
        av_lds[idx] = av * vv;
    }

    // ---- invariant A-fragments (t_aug) for the two M-tiles: 2 x 33 x v2f ----
    v2f afragA[33], afragB[33];
    const int mA = m0 + lm;
    const int mB = m0 + 16 + lm;
    #pragma unroll
    for (int s = 0; s < 33; ++s) {
        int k0 = 4 * s + kb;
        afragA[s].x = taug_elem(text, mA, k0);
        afragA[s].y = taug_elem(text, mA, k0 + 1);
        afragB[s].x = taug_elem(text, mB, k0);
        afragB[s].y = taug_elem(text, mB, k0 + 1);
    }

    __syncthreads();

    float accA[8], accB[8];
    #pragma unroll
    for (int r = 0; r < 8; ++r) { accA[r] = 0.0f; accB[r] = 0.0f; }

    for (int a = 0; a < AV_PER_SLICE; ++a) {
        const int avi = avi0 + a;
        // B column for this lane: W1[avi*129 + k][nb + lm], rows stride 128 floats
        const float* __restrict__ Wp =
            W1 + ((size_t)avi * KT + kb) * PF + nb + lm;

        v8f cA = {0.f, 0.f, 0.f, 0.f, 0.f, 0.f, 0.f, 0.f};
        v8f cB = {0.f, 0.f, 0.f, 0.f, 0.f, 0.f, 0.f, 0.f};
        #pragma unroll
        for (int s = 0; s < 32; ++s) {
            v2f b;
            b.x = Wp[0];
            b.y = Wp[PF];
            cA = __builtin_amdgcn_wmma_f32_16x16x4_f32(
                     false, afragA[s], false, b, (short)0, cA, false, false);
            cB = __builtin_amdgcn_wmma_f32_16x16x4_f32(
                     false, afragB[s], false, b, (short)0, cB, false, false);
            Wp += 4 * PF;
        }
        // tail step: k = 128..131; only k==128 valid (A zero-padded elsewhere).
        {
            const float w128 = W1[((size_t)avi * KT + (KT - 1)) * PF + nb + lm];
            v2f b; b.x = w128; b.y = w128;
            cA = __builtin_amdgcn_wmma_f32_16x16x4_f32(
                     false, afragA[32], false, b, (short)0, cA, false, false);
            cB = __builtin_amdgcn_wmma_f32_16x16x4_f32(
                     false, afragB[32], false, b, (short)0, cB, false, false);
        }
        // epilogue: scale by (a*v)[row, avi] (LDS broadcast within half-wave)
        #pragma unroll
        for (int r = 0; r < 8; ++r) {
            int rowA = r + 8 * hi;
            float scA = av_lds[rowA * AV_PER_SLICE + a];
            float scB = av_lds[(16 + rowA) * AV_PER_SLICE + a];
            accA[r] = fmaf(scA, cA[r], accA[r]);
            accB[r] = fmaf(scB, cB[r], accB[r]);
        }
    }

    // accumulate partial pre-activation h1 into workspace
    #pragma unroll
    for (int r = 0; r < 8; ++r) {
        int rowA = m0 + r + 8 * hi;
        atomicAdd(&hpre[rowA * PF + nb + lm], accA[r]);
        atomicAdd(&hpre[(rowA + 16) * PF + nb + lm], accB[r]);
    }
}

// Kernel 2: h1 = relu(hpre + b1); h2 = relu(h1@W2 + b2); out = h2@W3 + b3
__global__ __launch_bounds__(256) void tfn_head_kernel(
    const float* __restrict__ hpre, const float* __restrict__ b1,
    const float* __restrict__ W2,   const float* __restrict__ b2,
    const float* __restrict__ W3,   const float* __restrict__ b3,
    float* __restrict__ out)
{
    __shared__ float h2_lds[16 * PF];

    const int tid  = threadIdx.x;
    const int m0   = blockIdx.x * 16;
    const int lane = tid & 31;
    const int wave = tid >> 5;
    const int lm   = lane & 15;
    const int hi   = lane >> 4;
    const int kb   = hi * 2;
    const int nb   = wave * 16;

    // A-fragments: relu(hpre + b1) generated on the fly (K = 128 -> 32 steps)
    v2f af[32];
    const int m = m0 + lm;
    #pragma unroll
    for (int s = 0; s < 32; ++s) {
        int k0 = 4 * s + kb;
        af[s].x = fmaxf(hpre[m * PF + k0]     + b1[k0],     0.0f);
        af[s].y = fmaxf(hpre[m * PF + k0 + 1] + b1[k0 + 1], 0.0f);
    }

    v8f c = {0.f, 0.f, 0.f, 0.f, 0.f, 0.f, 0.f, 0.f};
    const float* __restrict__ Wp = W2 + kb * PF + nb + lm;
    #pragma unroll
    for (int s = 0; s < 32; ++s) {
        v2f b;
        b.x = Wp[0];
        b.y = Wp[PF];
        c = __builtin_amdgcn_wmma_f32_16x16x4_f32(
                false, af[s], false, b, (short)0, c, false, false);
        Wp += 4 * PF;
    }

    // h2 tile -> LDS with bias + relu
    const float b2v = b2[nb + lm];
    #pragma unroll
    for (int r = 0; r < 8; ++r) {
        h2_lds[(r + 8 * hi) * PF + nb + lm] = fmaxf(c[r] + b2v, 0.0f);
    }
    __syncthreads();

    // final dot with W3 (128,1) + b3
    if (tid < 16) {
        float acc = b3[0];
        #pragma unroll 8
        for (int j = 0; j < PF; ++j) acc = fmaf(h2_lds[tid * PF + j], W3[j], acc);
        out[m0 + tid] = acc;
    }
}

extern "C" void kernel_launch(void* const* d_in, const int* in_sizes, int n_in,
                              void* d_out, int out_size, void* d_ws, size_t ws_size,
                              hipStream_t stream) {
    const float* audio = (const float*)d_in[0];
    const float* video = (const float*)d_in[1];
    const float* text  = (const float*)d_in[2];
    const float* W1    = (const float*)d_in[3];
    const float* b1    = (const float*)d_in[4];
    const float* W2    = (const float*)d_in[5];
    const float* b2    = (const float*)d_in[6];
    const float* W3    = (const float*)d_in[7];
    const float* b3    = (const float*)d_in[8];
    float* out  = (float*)d_out;
    float* hpre = (float*)d_ws;   // 256*128 floats = 128 KB pre-activation accumulator

    hipMemsetAsync(hpre, 0, (size_t)BATCH * PF * sizeof(float), stream);

    dim3 grid1(BATCH / MROWS, SLICES);   // 8 M-slabs x 33 avi-slices = 264 workgroups
    tfn_w1_kernel<<<grid1, 256, 0, stream>>>(audio, video, text, W1, hpre);

    tfn_head_kernel<<<BATCH / 16, 256, 0, stream>>>(hpre, b1, W2, b2, W3, b3, out);
}